// SNN_26199300506034
// MI455X (gfx1250) — compile-verified
//
#include <hip/hip_runtime.h>

// ---------------------------------------------------------------------------
// SNN (3-layer LIF, snntorch-style) forward on MI455X (gfx1250, wave32).
// Exact f32 numerics via V_WMMA_F32_16X16X4_F32.
// Grid: 16 blocks (one per 16-row batch tile); block: 1024 threads = 32 waves.
// Wave w owns neurons [16w,16w+16) of layers 0/1; syn/mem state lives in the
// WMMA C-layout (v8f) registers for the whole 100-step recurrence.
// ---------------------------------------------------------------------------

typedef __attribute__((ext_vector_type(2))) float v2f;
typedef __attribute__((ext_vector_type(8))) float v8f;

#define ALPHA 0.5f
#define BETA  0.5f

constexpr int BSZ  = 256;
constexpr int DIN  = 700;
constexpr int HID  = 512;
constexpr int DOUT = 20;

// LDS row strides (floats). Chosen even (8B-aligned b64 loads) and with
// (stride mod 64) == 4 so the A-operand ds_load_b64 pattern
// (lane l -> row l&15, col k0 + 2*(l>>4)) touches all 64 banks conflict-free.
constexpr int XS = 708;   // x tile rows (>= 700)
constexpr int SS = 516;   // spike buffer rows (>= 512)

__device__ __forceinline__ v8f wmma_k4(v8f acc, v2f a, v2f b) {
    // D = A(16x4,f32) * B(4x16,f32) + C(16x16,f32)
    return __builtin_amdgcn_wmma_f32_16x16x4_f32(
        /*neg_a=*/false, a, /*neg_b=*/false, b,
        /*c_mod=*/(short)0, acc, /*reuse_a=*/false, /*reuse_b=*/false);
}

__global__ __launch_bounds__(1024, 1)
void snn_lif_wmma(const float* __restrict__ x,
                  const float* __restrict__ W0,
                  const float* __restrict__ W1,
                  const float* __restrict__ W2,
                  const float* __restrict__ th0p,
                  const float* __restrict__ th1p,
                  const float* __restrict__ th2p,
                  const int*   __restrict__ nsp,
                  float* __restrict__ out,
                  int Tx) {
    extern __shared__ float smem[];
    float* x_lds    = smem;                       // 16 * XS
    float* spk0_lds = x_lds    + 16 * XS;         // 16 * SS
    float* spk1_lds = spk0_lds + 16 * SS;         // 16 * SS
    float* cur2_lds = spk1_lds + 16 * SS;         // 16 * 32

    const int tid = threadIdx.x;
    const int w   = tid >> 5;          // wave id 0..31 (wave32)
    const int l15 = tid & 15;
    const int hi  = (tid >> 4) & 1;    // lane half (A/B K-pair select)
    const int b16 = blockIdx.x * 16;   // batch-row base
    const int n0  = w * 16;            // this wave's neuron tile (layers 0/1)

    const float TH0 = th0p[0];
    const float TH1 = th1p[0];
    const float TH2 = th2p[0];
    const int   NS  = nsp[0];

    // Output regions, flat in reference return order:
    // spk0, spk1, spk2, mem0, mem1, mem2
    const size_t rows  = (size_t)NS * BSZ;
    const size_t big   = rows * HID;
    const size_t small = rows * DOUT;
    const size_t O_SPK0 = 0;
    const size_t O_SPK1 = big;
    const size_t O_SPK2 = 2 * big;
    const size_t O_MEM0 = 2 * big + small;
    const size_t O_MEM1 = 3 * big + small;
    const size_t O_MEM2 = 4 * big + small;

    // Per-lane weight row pointers (B operand: B[k][n] = W[n][k]).
    // Lane l supplies column n0+l15; VGPR pair holds K = k0+2*hi, k0+2*hi+1.
    const float* w0row = W0 + (size_t)(n0 + l15) * DIN + 2 * hi;
    const float* w1row = W1 + (size_t)(n0 + l15) * HID + 2 * hi;

    // Layer 2: split-K across all 32 waves. Waves 0..15 -> N-tile 0 (n 0..15),
    // waves 16..31 -> N-tile 1 (n 16..31, only n<20 valid). Each wave covers
    // K chunk of 32.
    const int tt   = w >> 4;
    const int n0_2 = tt * 16;
    int r2 = n0_2 + l15; if (r2 > DOUT - 1) r2 = DOUT - 1;   // clamp OOB rows
    const float* w2row = W2 + (size_t)r2 * HID + 2 * hi;
    const int kb2 = (w & 15) * 32;

    const v8f zero = {0.f, 0.f, 0.f, 0.f, 0.f, 0.f, 0.f, 0.f};
    v8f syn0 = zero, mem0 = zero;
    v8f syn1 = zero, mem1 = zero;
    v8f syn2 = zero, mem2 = zero;   // live only in waves 0 and 16

    for (int t = 0; t < NS; ++t) {
        // ---- phase A: stage x[b16..b16+15, t, :] into LDS (cooperative) ----
        {
            const int row = tid >> 6;          // 16 rows x 64 threads
            const int c0  = tid & 63;
            const float* gx = x + ((size_t)(b16 + row) * Tx + t) * DIN;
            float* lx = x_lds + row * XS;
            for (int k = c0; k < DIN; k += 64) lx[k] = gx[k];
        }
        __syncthreads();                                   // B1

        // ---- phase B: layer 0 (K = 700 = 175 x 4) ----
        if (tid < 16 * 32) cur2_lds[tid] = 0.f;            // zero L2 accum
        v8f acc = zero;
        {
            const float* arow = x_lds + l15 * XS + 2 * hi;
            #pragma unroll 5
            for (int k = 0; k < DIN; k += 4) {
                v2f a = *(const v2f*)(arow + k);           // ds_load_b64
                v2f b = *(const v2f*)(w0row + k);          // global_load_b64
                acc = wmma_k4(acc, a, b);
            }
        }
        #pragma unroll
        for (int i = 0; i < 8; ++i) {                      // C layout: M=i+8*hi
            const int M = i + 8 * hi;
            float s     = ALPHA * syn0[i] + acc[i];
            float mprev = mem0[i];
            float rst   = (mprev > TH0) ? TH0 : 0.f;       // reset from PREV mem
            float m     = BETA * mprev + s - rst;
            float spk   = (m > TH0) ? 1.f : 0.f;           // Heaviside(mem - th)
            syn0[i] = s; mem0[i] = m;
            spk0_lds[M * SS + n0 + l15] = spk;
            const size_t ridx = ((size_t)t * BSZ + (b16 + M)) * HID + (n0 + l15);
            out[O_SPK0 + ridx] = spk;
            out[O_MEM0 + ridx] = m;
        }
        __syncthreads();                                   // B2

        // ---- phase C: layer 1 (K = 512 = 128 x 4) ----
        acc = zero;
        {
            const float* arow = spk0_lds + l15 * SS + 2 * hi;
            #pragma unroll 4
            for (int k = 0; k < HID; k += 4) {
                v2f a = *(const v2f*)(arow + k);
                v2f b = *(const v2f*)(w1row + k);
                acc = wmma_k4(acc, a, b);
            }
        }
        #pragma unroll
        for (int i = 0; i < 8; ++i) {
            const int M = i + 8 * hi;
            float s     = ALPHA * syn1[i] + acc[i];
            float mprev = mem1[i];
            float rst   = (mprev > TH1) ? TH1 : 0.f;
            float m     = BETA * mprev + s - rst;
            float spk   = (m > TH1) ? 1.f : 0.f;
            syn1[i] = s; mem1[i] = m;
            spk1_lds[M * SS + n0 + l15] = spk;
            const size_t ridx = ((size_t)t * BSZ + (b16 + M)) * HID + (n0 + l15);
            out[O_SPK1 + ridx] = spk;
            out[O_MEM1 + ridx] = m;
        }
        __syncthreads();                                   // B3

        // ---- phase D: layer 2 split-K partials (each wave: K chunk of 32) ----
        acc = zero;
        {
            const float* arow = spk1_lds + l15 * SS + kb2 + 2 * hi;
            const float* brow = w2row + kb2;
            #pragma unroll
            for (int k = 0; k < 32; k += 4) {
                v2f a = *(const v2f*)(arow + k);
                v2f b = *(const v2f*)(brow + k);
                acc = wmma_k4(acc, a, b);
            }
        }
        #pragma unroll
        for (int i = 0; i < 8; ++i) {
            const int M = i + 8 * hi;
            atomicAdd(&cur2_lds[M * 32 + n0_2 + l15], acc[i]);  // ds_add_f32
        }
        __syncthreads();                                   // B4

        // ---- phase E: waves 0 and 16 finalize layer-2 LIF + records ----
        if ((w & 15) == 0) {
            #pragma unroll
            for (int i = 0; i < 8; ++i) {
                const int M = i + 8 * hi;
                const int n = n0_2 + l15;
                float cur   = cur2_lds[M * 32 + n];
                float s     = ALPHA * syn2[i] + cur;
                float mprev = mem2[i];
                float rst   = (mprev > TH2) ? TH2 : 0.f;
                float m     = BETA * mprev + s - rst;
                float spk   = (m > TH2) ? 1.f : 0.f;
                syn2[i] = s; mem2[i] = m;
                if (n < DOUT) {
                    const size_t ridx = ((size_t)t * BSZ + (b16 + M)) * DOUT + n;
                    out[O_SPK2 + ridx] = spk;
                    out[O_MEM2 + ridx] = m;
                }
            }
        }
        // x_lds rewrite (next A) is fenced from this step's consumers by B2/B1;
        // spk buffers by B3/B4; cur2 zeroing (next B) fenced from E by B1.
    }
}

extern "C" void kernel_launch(void* const* d_in, const int* in_sizes, int n_in,
                              void* d_out, int out_size, void* d_ws, size_t ws_size,
                              hipStream_t stream) {
    const float* x   = (const float*)d_in[0];
    const float* W0  = (const float*)d_in[1];
    const float* W1  = (const float*)d_in[2];
    const float* W2  = (const float*)d_in[3];
    const float* th0 = (const float*)d_in[4];
    const float* th1 = (const float*)d_in[5];
    const float* th2 = (const float*)d_in[6];
    const int*   ns  = (const int*)d_in[7];
    float* out = (float*)d_out;

    const int Tx = in_sizes[0] / (BSZ * DIN);   // time length of x (=100)

    const size_t shmem = (size_t)(16 * XS + 2 * 16 * SS + 16 * 32) * sizeof(float);
    dim3 grid(BSZ / 16);      // 16 blocks, one per 16-row batch tile
    dim3 block(1024);         // 32 wave32 waves

    snn_lif_wmma<<<grid, block, shmem, stream>>>(x, W0, W1, W2, th0, th1, th2,
                                                 ns, out, Tx);
}